// FinalGraphTransformerModule_14508399526691
// MI455X (gfx1250) — compile-verified
//
#include <hip/hip_runtime.h>
#include <hip/hip_bf16.h>

// ---------------------------------------------------------------------------
// Graph Transformer layer for MI455X (gfx1250, wave32, WMMA f32 16x16x4,
// async global->LDS staging, prefetched L2 gathers)
// ---------------------------------------------------------------------------

typedef float v2f __attribute__((ext_vector_type(2)));
typedef float v8f __attribute__((ext_vector_type(8)));

__device__ __forceinline__ v8f wmma_f32(v2f a, v2f b, v8f c) {
  // D = A(16x4) * B(4x16) + C(16x16), fp32 throughout
  return __builtin_amdgcn_wmma_f32_16x16x4_f32(false, a, false, b, (short)0, c,
                                               false, false);
}

__device__ __forceinline__ float clip5(float x) {
  return fminf(fmaxf(x, -5.f), 5.f);
}

// CDNA5 async copy: 16 bytes/lane global -> LDS, tracked by ASYNCcnt.
// lds_off = byte offset within the workgroup LDS allocation (low 32 bits of a
// generic pointer to shared memory), gaddr = 16B-aligned global address.
__device__ __forceinline__ void async_g2lds_b128(unsigned lds_off,
                                                 const void* gaddr) {
  unsigned long long g = (unsigned long long)gaddr;
  asm volatile("global_load_async_to_lds_b128 %0, %1, off" ::"v"(lds_off),
               "v"(g)
               : "memory");
}

__device__ __forceinline__ void wait_async0() {
  asm volatile("s_wait_asynccnt 0x0" ::: "memory");
}

__device__ __forceinline__ unsigned lds_offset_of(const void* shared_ptr) {
  // generic pointer to LDS = {SHARED_BASE, offset}; low 32 bits = LDS offset
  return (unsigned)(unsigned long long)shared_ptr;
}

// ---------------------------------------------------------------------------
// 0) zero a float region
// ---------------------------------------------------------------------------
__global__ __launch_bounds__(256) void zero_kernel(float* __restrict__ p,
                                                   long n) {
  long i = (long)blockIdx.x * 256 + threadIdx.x;
  long stride = (long)gridDim.x * 256;
  for (; i < n; i += stride) p[i] = 0.f;
}

// ---------------------------------------------------------------------------
// 1) per-column sum / sum-of-squares over rows (D = 128 fixed)
// ---------------------------------------------------------------------------
__global__ __launch_bounds__(256) void col_stats_kernel(
    const float* __restrict__ X, long nrows, float* __restrict__ sum,
    float* __restrict__ sq) {
  const int t = threadIdx.x;
  const int col = t & 127;
  const int rsub = t >> 7;  // 2 row sub-streams per block
  float s = 0.f, q = 0.f;
  for (long r = (long)blockIdx.x * 2 + rsub; r < nrows;
       r += (long)gridDim.x * 2) {
    float v = X[r * 128 + col];
    s += v;
    q += v * v;
  }
  __shared__ float bs[256], bq[256];
  bs[t] = s;
  bq[t] = q;
  __syncthreads();
  if (t < 128) {
    atomicAdd(&sum[col], bs[t] + bs[t + 128]);
    atomicAdd(&sq[col], bq[t] + bq[t + 128]);
  }
}

// ---------------------------------------------------------------------------
// 2) fold BatchNorm into a weight matrix: Wf[k][j] = scale[k] * W[k][j]
// ---------------------------------------------------------------------------
__global__ __launch_bounds__(256) void fold_weight_kernel(
    const float* __restrict__ sum, const float* __restrict__ sq, float cnt,
    const float* __restrict__ gamma, const float* __restrict__ beta,
    const float* __restrict__ W, float* __restrict__ Wf, int J) {
  const int k = blockIdx.x;  // 0..127
  const float mean = sum[k] / cnt;
  const float var = sq[k] / cnt - mean * mean;
  const float scale = gamma[k] * rsqrtf(var + 1e-5f);
  for (int j = threadIdx.x; j < J; j += blockDim.x)
    Wf[k * J + j] = scale * W[k * J + j];
}

// bias[j] = sum_k shift[k] * W[k][j],  shift[k] = beta[k] - mean[k]*scale[k]
__global__ __launch_bounds__(256) void fold_bias_kernel(
    const float* __restrict__ sum, const float* __restrict__ sq, float cnt,
    const float* __restrict__ gamma, const float* __restrict__ beta,
    const float* __restrict__ W, float* __restrict__ biasOut, int J) {
  for (int j = threadIdx.x; j < J; j += blockDim.x) {
    float acc = 0.f;
    for (int k = 0; k < 128; ++k) {
      float mean = sum[k] / cnt;
      float var = sq[k] / cnt - mean * mean;
      float scale = gamma[k] * rsqrtf(var + 1e-5f);
      float shift = beta[k] - mean * scale;
      acc += shift * W[k * J + j];
    }
    biasOut[j] = acc;
  }
}

// ---------------------------------------------------------------------------
// 3) generic tiled WMMA GEMM:  out[M x J] = f(A[M x 128] @ W[128 x J])
//    block = 256 threads (8 waves), tile = 64 rows x 64 cols,
//    each wave owns two 16x16 output tiles (independent WMMA accum chains).
//    A-tile staged with GLOBAL_LOAD_ASYNC_TO_LDS_B128 unless DIVZ (which must
//    divide by z during staging). LDS pitch 132 floats: 16B-aligned rows +
//    conflict-free A-operand reads.
// ---------------------------------------------------------------------------
template <bool DIVZ, bool BIAS, bool SILU, bool RES, bool STATS>
__global__ __launch_bounds__(256) void gemm_tile_kernel(
    const float* __restrict__ A, const float* __restrict__ Zc,
    const float* __restrict__ W, const float* __restrict__ bias,
    const float* __restrict__ res, float* __restrict__ out,
    float* __restrict__ stat_sum, float* __restrict__ stat_sq, int M, int J,
    int lda, int aoff) {
  extern __shared__ float smem[];  // 64 x 132 floats (33.8 KB)
  __shared__ float colsum[64], colsq[64];
  const int t = threadIdx.x;
  const int lane = t & 31, wave = t >> 5;
  const int half = lane >> 4, l16 = lane & 15;
  const int m0 = blockIdx.x * 64;
  const int j0 = blockIdx.y * 64;
  constexpr int LDA = 132;

  if (STATS && t < 64) {
    colsum[t] = 0.f;
    colsq[t] = 0.f;
  }

  if (DIVZ) {
    // scalar staging (per-element divide by softmax denominator)
    for (int idx = t; idx < 64 * 128; idx += 256) {
      const int r = idx >> 7, k = idx & 127;
      const int row = m0 + r;
      float v = 0.f;
      if (row < M) {
        v = A[(size_t)row * lda + aoff + k];
        v = v / (Zc[(size_t)row * 4 + (k >> 5)] + 1e-6f);
      }
      smem[r * LDA + k] = v;
    }
  } else {
    // async staging: 64 rows x 32 x 16B chunks = 2048 chunks, 8 per thread
    const unsigned sbase = lds_offset_of(smem);
#pragma unroll
    for (int i = 0; i < 8; ++i) {
      const int idx = t + i * 256;
      const int r = idx >> 5, c = idx & 31;           // row, 16B chunk
      int row = m0 + r;
      row = (row < M) ? row : (M - 1);                // clamp: guarded rows
      const unsigned loff = sbase + (unsigned)(r * LDA + c * 4) * 4u;
      async_g2lds_b128(loff, A + (size_t)row * lda + aoff + c * 4);
    }
    wait_async0();
  }
  __syncthreads();

  const int rt = wave >> 1;        // row tile 0..3
  const int ct = (wave & 1) * 32;  // col offset 0 / 32 within 64-wide tile
  const float* Arow = &smem[(rt * 16 + l16) * LDA];
  const int c0 = j0 + ct + l16;
  const int c1 = c0 + 16;

  v8f acc0 = {0, 0, 0, 0, 0, 0, 0, 0};
  v8f acc1 = {0, 0, 0, 0, 0, 0, 0, 0};

#pragma unroll 8
  for (int k0 = 0; k0 < 128; k0 += 4) {
    const int ka = k0 + 2 * half;  // A: lanes 0-15 K=k0,k0+1; 16-31 K=k0+2,3
    v2f a;
    a.x = Arow[ka];
    a.y = Arow[ka + 1];
    const float* w0 = &W[(size_t)ka * J];
    const float* w1 = w0 + J;
    v2f b0, b1;
    b0.x = w0[c0];
    b0.y = w1[c0];
    b1.x = w0[c1];
    b1.y = w1[c1];
    acc0 = wmma_f32(a, b0, acc0);
    acc1 = wmma_f32(a, b1, acc1);
  }

  // epilogue: C/D layout -> lane l16 = column, VGPR v -> row v + 8*half
#pragma unroll
  for (int v = 0; v < 8; ++v) {
    const int row = m0 + rt * 16 + v + 8 * half;
    if (row < M) {
      float x0 = acc0[v];
      float x1 = acc1[v];
      if (BIAS) {
        x0 += bias[c0];
        x1 += bias[c1];
      }
      if (SILU) {
        x0 = x0 / (1.f + __expf(-x0));
        x1 = x1 / (1.f + __expf(-x1));
      }
      if (RES) {
        x0 += res[(size_t)row * J + c0];
        x1 += res[(size_t)row * J + c1];
      }
      out[(size_t)row * J + c0] = x0;
      out[(size_t)row * J + c1] = x1;
      if (STATS) {
        atomicAdd(&colsum[ct + l16], x0);
        atomicAdd(&colsq[ct + l16], x0 * x0);
        atomicAdd(&colsum[ct + l16 + 16], x1);
        atomicAdd(&colsq[ct + l16 + 16], x1 * x1);
      }
    }
  }
  if (STATS) {
    __syncthreads();
    if (t < 64) {
      atomicAdd(&stat_sum[j0 + t], colsum[t]);
      atomicAdd(&stat_sq[j0 + t], colsq[t]);
    }
  }
}

// ---------------------------------------------------------------------------
// 4) fused edge kernel: per 32-edge tile
//      prefetch Q/K/V gather rows (hide L2 latency behind the GEMM)
//      proj = ef @ We' + be      (WMMA, ef async-staged into LDS)
//      s[e,h] = exp(clip(sum_i clip(K[src]*Q[dst]/sqrt(32))*proj))
//      atomics: z[dst,h] += s ;  wV[dst,:] += s * V[src,:]
//    Q/K/V/wV/z all fit in the 192MB L2 -> gathers/atomics stay on-chip.
// ---------------------------------------------------------------------------
__global__ __launch_bounds__(256) void edge_attn_kernel(
    const float* __restrict__ EF, const int* __restrict__ src,
    const int* __restrict__ dst, const float* __restrict__ Wef,
    const float* __restrict__ be, const float* __restrict__ Q,
    const float* __restrict__ K, const float* __restrict__ V,
    float* __restrict__ wV, float* __restrict__ z, long Etot) {
  __shared__ float Ae[32 * 132];  // staged edge features (16B-aligned rows)
  __shared__ float Pe[32 * 132];  // projected edge features
  __shared__ int sidx[32], didx[32];
  __shared__ float sS[32 * 4];
  const int t = threadIdx.x;
  const int lane = t & 31, wave = t >> 5;
  const int half = lane >> 4, l16 = lane & 15;
  const long e0 = (long)blockIdx.x * 32;
  constexpr int LDA = 132;

  // prefetch the random gather rows now; GEMM below hides the latency
  if (t < 128) {
    const int e = t >> 2, h = t & 3;
    if (e0 + e < Etot) {
      const int se = src[e0 + e];
      const int de = dst[e0 + e];
      __builtin_prefetch(&K[(size_t)se * 128 + h * 32], 0, 3);
      __builtin_prefetch(&Q[(size_t)de * 128 + h * 32], 0, 3);
      __builtin_prefetch(&V[(size_t)se * 128 + h * 32], 0, 3);
      if (h == 0) {
        sidx[e] = se;
        didx[e] = de;
      }
    } else if (h == 0) {
      sidx[e] = 0;
      didx[e] = 0;
    }
  }

  // async-stage the 32x128 edge-feature tile (1024 x 16B chunks, 4/thread)
  {
    const unsigned sbase = lds_offset_of(Ae);
#pragma unroll
    for (int i = 0; i < 4; ++i) {
      const int idx = t + i * 256;
      const int r = idx >> 5, c = idx & 31;
      long e = e0 + r;
      e = (e < Etot) ? e : (Etot - 1);
      const unsigned loff = sbase + (unsigned)(r * LDA + c * 4) * 4u;
      async_g2lds_b128(loff, EF + e * 128 + c * 4);
    }
    wait_async0();
  }
  __syncthreads();

  // --- WMMA projection: 32x128 output = 2 row tiles x 8 col tiles, 8 waves
  {
    const int rt = wave & 1;
    const int cb = (wave >> 1) * 32;
    const float* Arow = &Ae[(rt * 16 + l16) * LDA];
    v8f acc0 = {0, 0, 0, 0, 0, 0, 0, 0};
    v8f acc1 = {0, 0, 0, 0, 0, 0, 0, 0};
#pragma unroll 8
    for (int k0 = 0; k0 < 128; k0 += 4) {
      const int ka = k0 + 2 * half;
      v2f a;
      a.x = Arow[ka];
      a.y = Arow[ka + 1];
      const float* w0 = &Wef[(size_t)ka * 128];
      const float* w1 = w0 + 128;
      v2f b0, b1;
      b0.x = w0[cb + l16];
      b0.y = w1[cb + l16];
      b1.x = w0[cb + 16 + l16];
      b1.y = w1[cb + 16 + l16];
      acc0 = wmma_f32(a, b0, acc0);
      acc1 = wmma_f32(a, b1, acc1);
    }
#pragma unroll
    for (int v = 0; v < 8; ++v) {
      const int r = rt * 16 + v + 8 * half;
      Pe[r * LDA + cb + l16] = acc0[v] + be[cb + l16];
      Pe[r * LDA + cb + 16 + l16] = acc1[v] + be[cb + 16 + l16];
    }
  }
  __syncthreads();

  // --- attention scores: one thread per (edge, head)
  if (t < 128) {
    const int e = t >> 2, h = t & 3;
    if (e0 + e < Etot) {
      const float4* kp = (const float4*)&K[(size_t)sidx[e] * 128 + h * 32];
      const float4* qp = (const float4*)&Q[(size_t)didx[e] * 128 + h * 32];
      const float* pp = &Pe[e * LDA + h * 32];
      const float rs = 0.17677669529663687f;  // 1/sqrt(32)
      float acc = 0.f;
#pragma unroll
      for (int i = 0; i < 8; ++i) {
        const float4 kv = kp[i];
        const float4 qv = qp[i];
        acc += clip5(kv.x * qv.x * rs) * pp[4 * i + 0];
        acc += clip5(kv.y * qv.y * rs) * pp[4 * i + 1];
        acc += clip5(kv.z * qv.z * rs) * pp[4 * i + 2];
        acc += clip5(kv.w * qv.w * rs) * pp[4 * i + 3];
      }
      const float s = __expf(clip5(acc));
      sS[t] = s;
      atomicAdd(&z[(size_t)didx[e] * 4 + h], s);
    }
  }
  __syncthreads();

  // --- scatter: wV[dst,:] += s * V[src,:]   (coalesced 128-float runs)
  for (int idx = t; idx < 32 * 128; idx += 256) {
    const int e = idx >> 7, d = idx & 127;
    if (e0 + e < Etot) {
      const float s = sS[e * 4 + (d >> 5)];
      atomicAdd(&wV[(size_t)didx[e] * 128 + d],
                s * V[(size_t)sidx[e] * 128 + d]);
    }
  }
}

// ---------------------------------------------------------------------------
// host launcher
// ---------------------------------------------------------------------------
extern "C" void kernel_launch(void* const* d_in, const int* in_sizes, int n_in,
                              void* d_out, int out_size, void* d_ws,
                              size_t ws_size, hipStream_t stream) {
  const float* node_feats = (const float*)d_in[0];
  const float* edge_feats = (const float*)d_in[1];
  const int* src = (const int*)d_in[2];
  const int* dst = (const int*)d_in[3];
  const float* Wq = (const float*)d_in[4];
  const float* Wk = (const float*)d_in[5];
  const float* Wv = (const float*)d_in[6];
  const float* We = (const float*)d_in[7];
  const float* O_w = (const float*)d_in[8];
  const float* O_b = (const float*)d_in[9];
  const float* g1n = (const float*)d_in[10];
  const float* b1n = (const float*)d_in[11];
  const float* g1e = (const float*)d_in[12];
  const float* b1e = (const float*)d_in[13];
  const float* g2 = (const float*)d_in[14];
  const float* b2 = (const float*)d_in[15];
  const float* W1 = (const float*)d_in[16];
  const float* W2 = (const float*)d_in[17];
  float* out = (float*)d_out;

  const long N = in_sizes[0] / 128;
  const long E = in_sizes[2];

  // ---- workspace layout (floats) ----
  float* ws = (float*)d_ws;
  size_t off = 0;
  auto alloc = [&](size_t n) {
    float* p = ws + off;
    off += n;
    return p;
  };
  float* wV = alloc((size_t)N * 128);
  float* zacc = alloc((size_t)N * 4);
  float* sum_n = alloc(128);
  float* sq_n = alloc(128);
  float* sum_e = alloc(128);
  float* sq_e = alloc(128);
  float* sum_h = alloc(128);
  float* sq_h = alloc(128);
  const size_t zero_n = off;  // everything above must start at 0
  float* Qm = alloc((size_t)N * 128);
  float* Km = alloc((size_t)N * 128);
  float* Vm = alloc((size_t)N * 128);
  float* h1 = alloc((size_t)N * 128);
  float* tmp = alloc((size_t)N * 256);
  float* Wqf = alloc(128 * 128);
  float* Wkf = alloc(128 * 128);
  float* Wvf = alloc(128 * 128);
  float* Wef = alloc(128 * 128);
  float* W1f = alloc(128 * 256);
  float* bq = alloc(128);
  float* bk = alloc(128);
  float* bv = alloc(128);
  float* be = alloc(128);
  float* b1 = alloc(256);

  const size_t shm = 64 * 132 * sizeof(float);

  // 0) zero accumulators + stats
  zero_kernel<<<2048, 256, 0, stream>>>(ws, (long)zero_n);

  // 1) BatchNorm statistics (node, edge)
  col_stats_kernel<<<512, 256, 0, stream>>>(node_feats, N, sum_n, sq_n);
  col_stats_kernel<<<2048, 256, 0, stream>>>(edge_feats, E, sum_e, sq_e);

  // 2) fold BN1 into Wq/Wk/Wv (node stats) and We (edge stats)
  fold_weight_kernel<<<128, 256, 0, stream>>>(sum_n, sq_n, (float)N, g1n, b1n,
                                              Wq, Wqf, 128);
  fold_weight_kernel<<<128, 256, 0, stream>>>(sum_n, sq_n, (float)N, g1n, b1n,
                                              Wk, Wkf, 128);
  fold_weight_kernel<<<128, 256, 0, stream>>>(sum_n, sq_n, (float)N, g1n, b1n,
                                              Wv, Wvf, 128);
  fold_weight_kernel<<<128, 256, 0, stream>>>(sum_e, sq_e, (float)E, g1e, b1e,
                                              We, Wef, 128);
  fold_bias_kernel<<<1, 256, 0, stream>>>(sum_n, sq_n, (float)N, g1n, b1n, Wq,
                                          bq, 128);
  fold_bias_kernel<<<1, 256, 0, stream>>>(sum_n, sq_n, (float)N, g1n, b1n, Wk,
                                          bk, 128);
  fold_bias_kernel<<<1, 256, 0, stream>>>(sum_n, sq_n, (float)N, g1n, b1n, Wv,
                                          bv, 128);
  fold_bias_kernel<<<1, 256, 0, stream>>>(sum_e, sq_e, (float)E, g1e, b1e, We,
                                          be, 128);

  // 3) Q/K/V projections (BN folded -> plain GEMM + bias)
  dim3 gN((unsigned)((N + 63) / 64), 2);
  gemm_tile_kernel<false, true, false, false, false>
      <<<gN, 256, shm, stream>>>(node_feats, nullptr, Wqf, bq, nullptr, Qm,
                                 nullptr, nullptr, (int)N, 128, 128, 0);
  gemm_tile_kernel<false, true, false, false, false>
      <<<gN, 256, shm, stream>>>(node_feats, nullptr, Wkf, bk, nullptr, Km,
                                 nullptr, nullptr, (int)N, 128, 128, 0);
  gemm_tile_kernel<false, true, false, false, false>
      <<<gN, 256, shm, stream>>>(node_feats, nullptr, Wvf, bv, nullptr, Vm,
                                 nullptr, nullptr, (int)N, 128, 128, 0);

  // 4) fused edge projection + attention + scatter
  edge_attn_kernel<<<(unsigned)((E + 31) / 32), 256, 0, stream>>>(
      edge_feats, src, dst, Wef, be, Qm, Km, Vm, wV, zacc, E);

  // 5) node update: h1 = node_feats + (wV / z) @ O_w + O_b, + BN2 stats
  gemm_tile_kernel<true, true, false, true, true>
      <<<gN, 256, shm, stream>>>(wV, zacc, O_w, O_b, node_feats, h1, sum_h,
                                 sq_h, (int)N, 128, 128, 0);

  // 6) fold BN2 into W1
  fold_weight_kernel<<<128, 256, 0, stream>>>(sum_h, sq_h, (float)N, g2, b2,
                                              W1, W1f, 256);
  fold_bias_kernel<<<1, 256, 0, stream>>>(sum_h, sq_h, (float)N, g2, b2, W1,
                                          b1, 256);

  // 7) MLP up: tmp = silu(h1 @ W1f + b1)   [N x 256]
  dim3 gU((unsigned)((N + 63) / 64), 4);
  gemm_tile_kernel<false, true, true, false, false>
      <<<gU, 256, shm, stream>>>(h1, nullptr, W1f, b1, nullptr, tmp, nullptr,
                                 nullptr, (int)N, 256, 128, 0);

  // 8) MLP down (K=256 split into two K=128 passes): out = h1 + tmp @ W2
  gemm_tile_kernel<false, false, false, true, false>
      <<<gN, 256, shm, stream>>>(tmp, nullptr, W2, nullptr, h1, out, nullptr,
                                 nullptr, (int)N, 128, 256, 0);
  gemm_tile_kernel<false, false, false, true, false>
      <<<gN, 256, shm, stream>>>(tmp, nullptr, W2 + 128 * 128, nullptr, out,
                                 out, nullptr, nullptr, (int)N, 128, 256, 128);
}